// QuanvolutionFilter_65481071410174
// MI455X (gfx1250) — compile-verified
//
#include <hip/hip_runtime.h>
#include <math.h>

typedef float v2f __attribute__((ext_vector_type(2)));
typedef float v8f __attribute__((ext_vector_type(8)));

// ============================================================================
// Setup kernel (1 block, 256 threads): build the fixed 16x16 circuit unitary U
// from ansatz_params, fold it with W into the four 16x16 quadratic-form
// matrices M_i, and emit G in WMMA-B-operand-swizzled order:
//   gswz[chunk][lane] = { G[4*chunk + 2*(lane/16) + 0][lane%16],
//                         G[4*chunk + 2*(lane/16) + 1][lane%16] }
// where G[f][i] = M_i[f>>4][f&15] for i<4, 0 for i>=4.  (64 chunks = K=256)
// ============================================================================
__global__ __launch_bounds__(256) void quanv_setup(
    const float* __restrict__ params,   // (3,4,3)
    const float* __restrict__ W,        // (4,4) row-major W[i][w]
    float2* __restrict__ gswz)          // [64][32] float2 = 16 KB
{
  __shared__ float sUre[16][16];        // sUre[k][j] = Re U[k][j]
  __shared__ float sUim[16][16];
  __shared__ float sW[16];
  const int t = threadIdx.x;
  if (t < 16) sW[t] = W[t];

  if (t < 16) {
    // Evolve basis state |t> through the fixed part of the circuit.
    // Wire w corresponds to bit (3-w) of the flat index (axis 1 = wire 0 = MSB).
    float ax[16], ay[16];
#pragma unroll
    for (int j = 0; j < 16; ++j) { ax[j] = (j == t) ? 1.f : 0.f; ay[j] = 0.f; }

#pragma unroll
    for (int l = 0; l < 3; ++l) {
      // Rot(phi,theta,omega) on each wire
#pragma unroll
      for (int w = 0; w < 4; ++w) {
        const float phi = params[(l * 4 + w) * 3 + 0];
        const float th  = params[(l * 4 + w) * 3 + 1];
        const float om  = params[(l * 4 + w) * 3 + 2];
        float st, ct;  sincosf(0.5f * th, &st, &ct);
        float shs, chs; sincosf(0.5f * (phi + om), &shs, &chs);
        float shd, chd; sincosf(0.5f * (phi - om), &shd, &chd);
        const float m00x =  ct * chs, m00y = -ct * shs;
        const float m01x = -st * chd, m01y = -st * shd;
        const float m10x =  st * chd, m10y = -st * shd;
        const float m11x =  ct * chs, m11y =  ct * shs;
        const int mask = 1 << (3 - w);
#pragma unroll
        for (int idx = 0; idx < 16; ++idx) {
          if (idx & mask) continue;
          const int pr = idx | mask;
          const float arx = ax[idx], ary = ay[idx];
          const float brx = ax[pr],  bry = ay[pr];
          ax[idx] = m00x * arx - m00y * ary + m01x * brx - m01y * bry;
          ay[idx] = m00x * ary + m00y * arx + m01x * bry + m01y * brx;
          ax[pr]  = m10x * arx - m10y * ary + m11x * brx - m11y * bry;
          ay[pr]  = m10x * ary + m10y * arx + m11x * bry + m11y * brx;
        }
      }
      // CNOT ring, range r = (l % 3) + 1
      const int r = (l % 3) + 1;
#pragma unroll
      for (int w = 0; w < 4; ++w) {
        const int tg = (w + r) & 3;
        const int cm = 1 << (3 - w), tm = 1 << (3 - tg);
#pragma unroll
        for (int idx = 0; idx < 16; ++idx) {
          if ((idx & cm) && !(idx & tm)) {
            const int pr = idx | tm;
            const float tx = ax[idx], ty = ay[idx];
            ax[idx] = ax[pr]; ay[idx] = ay[pr];
            ax[pr] = tx;      ay[pr] = ty;
          }
        }
      }
    }
#pragma unroll
    for (int k = 0; k < 16; ++k) { sUre[k][t] = ax[k]; sUim[k][t] = ay[k]; }
  }
  __syncthreads();

  // G[f][i] = M_i[j1][j2],  f = j1*16+j2,
  // M_i[j1][j2] = sum_k c_ik * (Ure[k][j1]*Ure[k][j2] + Uim[k][j1]*Uim[k][j2])
  // c_ik = sum_w W[i][w] * (1 - 2*bit_{3-w}(k))
  for (int rep = 0; rep < 8; ++rep) {
    const int idx  = rep * 256 + t;     // 0..2047
    const int cch  = idx >> 5;
    const int lane = idx & 31;
    const int hi   = lane >> 4;
    const int col  = lane & 15;
    float vv[2] = {0.f, 0.f};
    if (col < 4) {
      for (int q = 0; q < 2; ++q) {
        const int f  = 4 * cch + 2 * hi + q;
        const int j1 = f >> 4, j2 = f & 15;
        float acc = 0.f;
        for (int k = 0; k < 16; ++k) {
          float cik = 0.f;
#pragma unroll
          for (int w = 0; w < 4; ++w) {
            const float z = ((k >> (3 - w)) & 1) ? -1.f : 1.f;
            cik += sW[col * 4 + w] * z;
          }
          acc += cik * (sUre[k][j1] * sUre[k][j2] + sUim[k][j1] * sUim[k][j2]);
        }
        vv[q] = acc;
      }
    }
    gswz[idx] = make_float2(vv[0], vv[1]);
  }
}

// ============================================================================
// Main kernel: each wave handles 16 patches per tile via a chain of 64
// V_WMMA_F32_16X16X4_F32 ops computing  out_tile = Phi(16x256) x G(256x16).
// ============================================================================
__global__ __launch_bounds__(256) void quanv_main(
    const float* __restrict__ x,        // (B,1,28,28)
    const float* __restrict__ bvec,     // (4,)
    const float2* __restrict__ gswz,    // [64][32] float2
    float* __restrict__ out,            // (B, 784) flat = patch*4 + i
    int n_tiles)
{
  __shared__ float2 sG[2048];           // 16 KB staged B-operand table
  const int tid = threadIdx.x;
  for (int i = tid; i < 2048; i += 256) sG[i] = gswz[i];
  __syncthreads();

  const int lane = tid & 31;
  const int wv   = tid >> 5;
  const int hi   = lane >> 4;           // half-wave select
  const int n    = lane & 15;           // A-row (patch-in-tile) / D-col index
  const float bias = (n < 4) ? bvec[n] : 0.f;

  const int gw     = blockIdx.x * 8 + wv;
  const int stride = gridDim.x * 8;

  for (int tile = gw; tile < n_tiles; tile += stride) {
    const int p0 = tile << 4;
    // This lane's patch (A-matrix row n): decode (image, 2x2 patch position)
    const int p  = p0 + n;
    const int b  = p / 196;
    const int rc = p - b * 196;
    const int rr = rc / 14;
    const int cc = rc - rr * 14;
    const float* ip = x + (size_t)b * 784 + (rr * 2) * 28 + cc * 2;
    const float2 top = *(const float2*)(ip);        // img[2r,2c], img[2r,2c+1]
    const float2 bot = *(const float2*)(ip + 28);   // img[2r+1,2c], img[2r+1,2c+1]

    float s0, c0, s1, c1, s2, c2, s3, c3;
    __sincosf(0.5f * top.x, &s0, &c0);
    __sincosf(0.5f * top.y, &s1, &c1);
    __sincosf(0.5f * bot.x, &s2, &c2);
    __sincosf(0.5f * bot.y, &s3, &c3);

    // psi[j] = prod over wires of cos/sin, wire 0 = MSB of j
    const float p01[4] = {c0 * c1, c0 * s1, s0 * c1, s0 * s1};
    const float p23[4] = {c2 * c3, c2 * s3, s2 * c3, s2 * s3};
    float psi[16];
#pragma unroll
    for (int j = 0; j < 16; ++j) psi[j] = p01[j >> 2] * p23[j & 3];

    // Second outer-product factor, pre-selected by half-wave (hi):
    // u[2g+q] = psi[4g + 2*hi + q]
    float u[8];
#pragma unroll
    for (int g = 0; g < 4; ++g) {
      u[2 * g + 0] = hi ? psi[4 * g + 2] : psi[4 * g + 0];
      u[2 * g + 1] = hi ? psi[4 * g + 3] : psi[4 * g + 1];
    }

    v8f acc = {0.f, 0.f, 0.f, 0.f, 0.f, 0.f, 0.f, 0.f};
#pragma unroll
    for (int ch = 0; ch < 64; ++ch) {
      // A rows f = 4*ch + 2*hi + {0,1}:  phi_f = psi[f>>4] * psi[f&15]
      // (f>>4 == ch>>2 for both halves: no carry since (4*ch)%16 <= 12)
      const float first = psi[ch >> 2];
      v2f a;
      a.x = first * u[(ch & 3) * 2 + 0];
      a.y = first * u[(ch & 3) * 2 + 1];
      const float2 bg = sG[(ch << 5) + lane];   // conflict-free ds_load_b64
      v2f bb; bb.x = bg.x; bb.y = bg.y;
      acc = __builtin_amdgcn_wmma_f32_16x16x4_f32(
          false, a, false, bb, (short)0, acc, false, false);
    }

    // D layout: VGPR v, lane L -> D[v + 8*hi, L%16]; only cols n<4 are real.
    if (n < 4) {
      const int pbase = p0 + 8 * hi;
#pragma unroll
      for (int v = 0; v < 8; ++v)
        out[(size_t)(pbase + v) * 4 + n] = acc[v] + bias;
    }
  }
}

extern "C" void kernel_launch(void* const* d_in, const int* in_sizes, int n_in,
                              void* d_out, int out_size, void* d_ws, size_t ws_size,
                              hipStream_t stream) {
  const float* x      = (const float*)d_in[0];   // (B,1,28,28)
  const float* params = (const float*)d_in[1];   // (3,4,3)
  const float* W      = (const float*)d_in[2];   // (4,4)
  const float* b      = (const float*)d_in[3];   // (4,)
  float* out  = (float*)d_out;
  float2* gswz = (float2*)d_ws;                  // 16 KB used

  const int B = in_sizes[0] / 784;
  const int n_tiles = (B * 196) / 16;            // B=4096 -> 50176, exact

  hipLaunchKernelGGL(quanv_setup, dim3(1), dim3(256), 0, stream,
                     params, W, gswz);

  int blocks = 784;                               // 8 waves/block, ~8 tiles/wave
  if (blocks * 8 > n_tiles) blocks = (n_tiles + 7) / 8;
  if (blocks < 1) blocks = 1;
  hipLaunchKernelGGL(quanv_main, dim3(blocks), dim3(256), 0, stream,
                     x, b, (const float2*)gswz, out, n_tiles);
}